// MSGAT_71244917506207
// MI455X (gfx1250) — compile-verified
//
#include <hip/hip_runtime.h>
#include <hip/hip_bf16.h>

// ---------------------------------------------------------------------------
// Sizes (fixed by the reference)
// ---------------------------------------------------------------------------
#define NN 50000      // nodes
#define EE 500000     // edges (original)
#define EP 550000     // edges + N self loops
#define GG 64         // graphs
#define H1 4          // gat1 heads
#define CC 64         // per-head dim
#define D1 256        // H1*CC
#define NEG_SLOPE 0.2f

typedef float v2f __attribute__((ext_vector_type(2)));
typedef float v8f __attribute__((ext_vector_type(8)));

// ---- monotonic float<->uint mapping for atomic float max -------------------
__device__ __forceinline__ unsigned f2ord(float f) {
    unsigned u = __float_as_uint(f);
    return (u & 0x80000000u) ? ~u : (u | 0x80000000u);
}
__device__ __forceinline__ float ord2f(unsigned u) {
    unsigned v = (u & 0x80000000u) ? (u & 0x7FFFFFFFu) : ~u;
    return __uint_as_float(v);
}

__device__ __forceinline__ void edge_info(int e, const int* ei, const float* ea,
                                          const float* mattr,
                                          int& s, int& d, float& a0, float& a1) {
    if (e < EE) {
        s = ei[e]; d = ei[EE + e];
        a0 = ea[2 * e]; a1 = ea[2 * e + 1];
    } else {
        s = d = e - EE;
        a0 = mattr[2 * (e - EE)]; a1 = mattr[2 * (e - EE) + 1];
    }
}

// ---------------------------------------------------------------------------
// Generic zero fill (grid-stride)
// ---------------------------------------------------------------------------
__global__ void k_zero(float* p, size_t n) {
    size_t i = (size_t)blockIdx.x * blockDim.x + threadIdx.x;
    size_t stride = (size_t)gridDim.x * blockDim.x;
    for (; i < n; i += stride) p[i] = 0.0f;
}

// ---------------------------------------------------------------------------
// Self-loop attr: degree + attr sums per destination, then mean
// ---------------------------------------------------------------------------
__global__ void k_deg_attr(const int* ei, const float* ea, float* deg, float* mattr) {
    int e = blockIdx.x * blockDim.x + threadIdx.x;
    if (e >= EE) return;
    int dst = ei[EE + e];
    atomicAdd(&deg[dst], 1.0f);
    atomicAdd(&mattr[2 * dst + 0], ea[2 * e + 0]);
    atomicAdd(&mattr[2 * dst + 1], ea[2 * e + 1]);
}
__global__ void k_mean_attr(float* deg, float* mattr) {
    int n = blockIdx.x * blockDim.x + threadIdx.x;
    if (n >= NN) return;
    float inv = 1.0f / fmaxf(deg[n], 1.0f);
    mattr[2 * n + 0] *= inv;
    mattr[2 * n + 1] *= inv;
}

// ---------------------------------------------------------------------------
// GAT1 linears: K=5 is tiny -> memory bound VALU kernel. one thread per (n,c)
// ---------------------------------------------------------------------------
__global__ void k_lin1(const float* x, const float* Wl, const float* bl,
                       const float* Wr, const float* br,
                       float* xl, float* xr) {
    int i = blockIdx.x * blockDim.x + threadIdx.x;
    if (i >= NN * D1) return;
    int n = i >> 8, c = i & 255;
    float xv[5];
#pragma unroll
    for (int k = 0; k < 5; k++) xv[k] = x[n * 5 + k];
    float sl = bl[c], sr = br[c];
#pragma unroll
    for (int k = 0; k < 5; k++) {
        sl += xv[k] * Wl[k * D1 + c];
        sr += xv[k] * Wr[k * D1 + c];
    }
    xl[i] = sl;
    xr[i] = sr;
}

// ---------------------------------------------------------------------------
// GAT1: per-edge logits + segment max.  One wave32 per edge, 8 elems/lane,
// per-head (8-lane) shuffle reduction.
// ---------------------------------------------------------------------------
__global__ void k_logits1(const int* ei, const float* ea, const float* mattr,
                          const float* xl, const float* xr,
                          const float* We, const float* att,
                          float* elog, unsigned* nmax) {
    int lane = threadIdx.x & 31;
    int e = (blockIdx.x * blockDim.x + threadIdx.x) >> 5;
    if (e >= EP) return;
    int s, d; float a0, a1;
    edge_info(e, ei, ea, mattr, s, d, a0, a1);
    const float* xls = xl + (size_t)s * D1;
    const float* xrd = xr + (size_t)d * D1;
    int base = lane * 8;
    float part = 0.0f;
#pragma unroll
    for (int j = 0; j < 8; j++) {
        int i = base + j;
        float m = xls[i] + xrd[i] + a0 * We[i] + a1 * We[D1 + i];
        m = (m > 0.0f) ? m : m * NEG_SLOPE;
        part += m * att[i];   // att flat [H1*CC], index matches element index
    }
    part += __shfl_xor(part, 1, 32);
    part += __shfl_xor(part, 2, 32);
    part += __shfl_xor(part, 4, 32);
    if ((lane & 7) == 0) {
        int h = lane >> 3;
        elog[(size_t)e * H1 + h] = part;
        atomicMax(&nmax[(size_t)d * H1 + h], f2ord(part));
    }
}

__global__ void k_expsum1(const int* ei, const unsigned* nmax,
                          float* elog, float* nsum) {
    int idx = blockIdx.x * blockDim.x + threadIdx.x;
    if (idx >= EP * H1) return;
    int e = idx >> 2, h = idx & 3;
    int d = (e < EE) ? ei[EE + e] : (e - EE);
    float m = ord2f(nmax[(size_t)d * H1 + h]);
    float ex = expf(elog[idx] - m);
    elog[idx] = ex;
    atomicAdd(&nsum[(size_t)d * H1 + h], ex);
}

__global__ void k_agg1(const int* ei, const float* ea, const float* mattr,
                       const float* xl, const float* elog, const float* nsum,
                       float* acc) {
    int lane = threadIdx.x & 31;
    int e = (blockIdx.x * blockDim.x + threadIdx.x) >> 5;
    if (e >= EP) return;
    int s, d; float a0, a1;
    edge_info(e, ei, ea, mattr, s, d, a0, a1);
    int h = lane >> 3;
    float alpha = elog[(size_t)e * H1 + h] / nsum[(size_t)d * H1 + h];
    const float* xls = xl + (size_t)s * D1;
    float* accd = acc + (size_t)d * D1;
    int base = lane * 8;
#pragma unroll
    for (int j = 0; j < 8; j++) {
        int i = base + j;
        atomicAdd(&accd[i], xls[i] * alpha);
    }
}

__global__ void k_finish1(float* acc, const float* bias) {
    int i = blockIdx.x * blockDim.x + threadIdx.x;
    if (i >= NN * D1) return;
    float v = acc[i] + bias[i & 255];
    acc[i] = (v > 0.0f) ? v : (expf(v) - 1.0f);   // ELU
}

// ---------------------------------------------------------------------------
// GAT2 linears with WMMA: xl2/xr2[N,64] = h1[N,256] @ {Wl2,Wr2}[256,64] + b
//
// Weights (2 x 64KB) are staged into LDS ONCE per workgroup with the CDNA5
// async global->LDS copy path (GLOBAL_LOAD_ASYNC_TO_LDS_B128, ASYNCcnt),
// instead of every wave re-streaming 128KB from L2. Then one wave per 16-row
// tile feeds V_WMMA_F32_16X16X4_F32 with B fragments from ds_load.
// N = 50000 = 3125 * 16 exactly, so EXEC is all-ones in every active wave.
// ---------------------------------------------------------------------------
__global__ void k_lin2_wmma(const float* h1, const float* Wl, const float* bl,
                            const float* Wr, const float* br,
                            float* xl2, float* xr2) {
    extern __shared__ float lds[];           // 32768 floats = 128KB dynamic LDS
    float* lWl = lds;                        // [256*64]
    float* lWr = lds + D1 * CC;              // [256*64]

    // ---- async stage both weight matrices into LDS (b128 granularity) ----
    // 2 * 16384 floats = 8192 float4 chunks; 256 threads -> 32 chunks each.
    for (int v4 = threadIdx.x; v4 < 2 * D1 * CC / 4; v4 += blockDim.x) {
        const float* src = (v4 < D1 * CC / 4) ? (Wl + (size_t)v4 * 4)
                                              : (Wr + (size_t)(v4 - D1 * CC / 4) * 4);
        unsigned lds_off = (unsigned)(uintptr_t)(&lds[v4 * 4]);  // LDS byte addr
        asm volatile("global_load_async_to_lds_b128 %0, %1, off"
                     :: "v"(lds_off), "v"(src) : "memory");
    }
    asm volatile("s_wait_asynccnt 0x0" ::: "memory");
    __syncthreads();

    int wave = threadIdx.x >> 5;
    int lane = threadIdx.x & 31;
    int tile = blockIdx.x * (blockDim.x >> 5) + wave;
    if (tile * 16 >= NN) return;

    int mlocal = lane & 15;
    int khalf = (lane >> 4) * 2;           // lanes 16-31 carry K+2,K+3
    int row = tile * 16 + mlocal;
    const float* arow = h1 + (size_t)row * D1;

    v8f z8 = {0.f, 0.f, 0.f, 0.f, 0.f, 0.f, 0.f, 0.f};
    v8f accl[4] = {z8, z8, z8, z8};
    v8f accr[4] = {z8, z8, z8, z8};

    for (int k = 0; k < D1; k += 4) {
        v2f a;
        a.x = arow[k + khalf];
        a.y = arow[k + khalf + 1];
#pragma unroll
        for (int t = 0; t < 4; t++) {
            int ncol = t * 16 + mlocal;
            v2f bw, cw;
            bw.x = lWl[(k + khalf) * CC + ncol];
            bw.y = lWl[(k + khalf + 1) * CC + ncol];
            cw.x = lWr[(k + khalf) * CC + ncol];
            cw.y = lWr[(k + khalf + 1) * CC + ncol];
            accl[t] = __builtin_amdgcn_wmma_f32_16x16x4_f32(
                false, a, false, bw, (short)0, accl[t], false, false);
            accr[t] = __builtin_amdgcn_wmma_f32_16x16x4_f32(
                false, a, false, cw, (short)0, accr[t], false, false);
        }
    }

    // D layout: VGPR i -> M = (lane<16 ? i : 8+i), N = lane%16
    int mbase = (lane >> 4) * 8;
#pragma unroll
    for (int t = 0; t < 4; t++) {
        int ncol = t * 16 + mlocal;
#pragma unroll
        for (int i = 0; i < 8; i++) {
            int r = tile * 16 + mbase + i;
            xl2[(size_t)r * CC + ncol] = accl[t][i] + bl[ncol];
            xr2[(size_t)r * CC + ncol] = accr[t][i] + br[ncol];
        }
    }
}

// ---------------------------------------------------------------------------
// GAT2 edge kernels (1 head, C=64): wave per edge, 2 elems/lane
// ---------------------------------------------------------------------------
__global__ void k_logits2(const int* ei, const float* ea, const float* mattr,
                          const float* xl, const float* xr,
                          const float* We, const float* att,
                          float* elog, unsigned* nmax) {
    int lane = threadIdx.x & 31;
    int e = (blockIdx.x * blockDim.x + threadIdx.x) >> 5;
    if (e >= EP) return;
    int s, d; float a0, a1;
    edge_info(e, ei, ea, mattr, s, d, a0, a1);
    const float* xls = xl + (size_t)s * CC;
    const float* xrd = xr + (size_t)d * CC;
    float part = 0.0f;
#pragma unroll
    for (int j = 0; j < 2; j++) {
        int i = lane * 2 + j;
        float m = xls[i] + xrd[i] + a0 * We[i] + a1 * We[CC + i];
        m = (m > 0.0f) ? m : m * NEG_SLOPE;
        part += m * att[i];
    }
    part += __shfl_xor(part, 1, 32);
    part += __shfl_xor(part, 2, 32);
    part += __shfl_xor(part, 4, 32);
    part += __shfl_xor(part, 8, 32);
    part += __shfl_xor(part, 16, 32);
    if (lane == 0) {
        elog[e] = part;
        atomicMax(&nmax[d], f2ord(part));
    }
}

__global__ void k_expsum2(const int* ei, const unsigned* nmax,
                          float* elog, float* nsum) {
    int e = blockIdx.x * blockDim.x + threadIdx.x;
    if (e >= EP) return;
    int d = (e < EE) ? ei[EE + e] : (e - EE);
    float ex = expf(elog[e] - ord2f(nmax[d]));
    elog[e] = ex;
    atomicAdd(&nsum[d], ex);
}

__global__ void k_agg2(const int* ei, const float* ea, const float* mattr,
                       const float* xl, const float* elog, const float* nsum,
                       float* acc) {
    int lane = threadIdx.x & 31;
    int e = (blockIdx.x * blockDim.x + threadIdx.x) >> 5;
    if (e >= EP) return;
    int s, d; float a0, a1;
    edge_info(e, ei, ea, mattr, s, d, a0, a1);
    float alpha = elog[e] / nsum[d];
    const float* xls = xl + (size_t)s * CC;
    float* accd = acc + (size_t)d * CC;
#pragma unroll
    for (int j = 0; j < 2; j++) {
        int i = lane * 2 + j;
        atomicAdd(&accd[i], xls[i] * alpha);
    }
}

__global__ void k_finish2(float* acc, const float* bias) {
    int i = blockIdx.x * blockDim.x + threadIdx.x;
    if (i >= NN * CC) return;
    float v = acc[i] + bias[i & 63];
    acc[i] = (v > 0.0f) ? v : (expf(v) - 1.0f);
}

// ---------------------------------------------------------------------------
// Global mean pool (sum via atomics; divide at GRU input)
// ---------------------------------------------------------------------------
__global__ void k_pool(const int* batch, const float* h2, float* pool, float* cnt) {
    int i = blockIdx.x * blockDim.x + threadIdx.x;
    if (i >= NN * CC) return;
    int n = i >> 6, c = i & 63;
    int g = batch[n];
    atomicAdd(&pool[(size_t)g * CC + c], h2[i]);
    if (c == 0) atomicAdd(&cnt[g], 1.0f);
}

// ---------------------------------------------------------------------------
// GRU over G=64 pooled vectors (batch=1) + classifier. Single 64-thread block.
// ---------------------------------------------------------------------------
__global__ void k_gru_head(const float* pool, const float* cnt,
                           const float* Wi, const float* Wh,
                           const float* bi, const float* bh,
                           const float* Wc1, const float* bc1,
                           const float* Wc2, const float* bc2,
                           float* out) {
    __shared__ float h[CC], xt[CC], zb[32];
    int t = threadIdx.x;
    h[t] = 0.0f;
    __syncthreads();
    for (int step = 0; step < GG; step++) {
        xt[t] = pool[step * CC + t] / fmaxf(cnt[step], 1.0f);
        __syncthreads();
        float ir = bi[t], iz = bi[CC + t], in_ = bi[2 * CC + t];
        float hr = bh[t], hz = bh[CC + t], hn = bh[2 * CC + t];
        for (int k = 0; k < CC; k++) {
            float xv = xt[k], hv = h[k];
            ir += xv * Wi[k * 3 * CC + t];
            iz += xv * Wi[k * 3 * CC + CC + t];
            in_ += xv * Wi[k * 3 * CC + 2 * CC + t];
            hr += hv * Wh[k * 3 * CC + t];
            hz += hv * Wh[k * 3 * CC + CC + t];
            hn += hv * Wh[k * 3 * CC + 2 * CC + t];
        }
        float r = 1.0f / (1.0f + expf(-(ir + hr)));
        float z = 1.0f / (1.0f + expf(-(iz + hz)));
        float nn = tanhf(in_ + r * hn);
        float hnew = (1.0f - z) * nn + z * h[t];
        __syncthreads();
        h[t] = hnew;
        __syncthreads();
    }
    if (t < 32) {
        float a = bc1[t];
        for (int k = 0; k < CC; k++) a += h[k] * Wc1[k * 32 + t];
        zb[t] = fmaxf(a, 0.0f);
    }
    __syncthreads();
    if (t == 0) {
        float a = bc2[0];
        for (int k = 0; k < 32; k++) a += zb[k] * Wc2[k];
        out[0] = 1.0f / (1.0f + expf(-a));
    }
}

// ---------------------------------------------------------------------------
// Host launcher
// ---------------------------------------------------------------------------
extern "C" void kernel_launch(void* const* d_in, const int* in_sizes, int n_in,
                              void* d_out, int out_size, void* d_ws, size_t ws_size,
                              hipStream_t stream) {
    const float* x     = (const float*)d_in[0];
    const float* eattr = (const float*)d_in[1];
    const int*   eidx  = (const int*)d_in[2];
    const int*   batch = (const int*)d_in[3];
    const float* Wl1 = (const float*)d_in[4],  *bl1 = (const float*)d_in[5];
    const float* Wr1 = (const float*)d_in[6],  *br1 = (const float*)d_in[7];
    const float* We1 = (const float*)d_in[8],  *att1 = (const float*)d_in[9];
    const float* bias1 = (const float*)d_in[10];
    const float* Wl2 = (const float*)d_in[11], *bl2 = (const float*)d_in[12];
    const float* Wr2 = (const float*)d_in[13], *br2 = (const float*)d_in[14];
    const float* We2 = (const float*)d_in[15], *att2 = (const float*)d_in[16];
    const float* bias2 = (const float*)d_in[17];
    const float* Wi = (const float*)d_in[18], *Wh = (const float*)d_in[19];
    const float* bi = (const float*)d_in[20], *bh = (const float*)d_in[21];
    const float* Wc1 = (const float*)d_in[22], *bc1 = (const float*)d_in[23];
    const float* Wc2 = (const float*)d_in[24], *bc2 = (const float*)d_in[25];
    float* out = (float*)d_out;

    // ---- workspace layout (floats) ----
    float* w = (float*)d_ws;
    size_t o = 0;
    float* xl1  = w + o; o += (size_t)NN * D1;     // also reused as xl2
    float* xr1  = w + o; o += (size_t)NN * D1;     // also reused as xr2
    float* acc1 = w + o; size_t zstart = o; o += (size_t)NN * D1;  // out1 -> h1
    float* elog = w + o; o += (size_t)EP * H1;     // logits/ex (GAT2 uses first EP)
    float* nmaxf = w + o; o += (size_t)NN * H1;    // ord-encoded max
    float* nsum = w + o; o += (size_t)NN * H1;
    float* deg  = w + o; o += (size_t)NN;
    float* mattr = w + o; o += (size_t)NN * 2;
    float* acc2 = w + o; o += (size_t)NN * CC;     // out2 -> h2
    float* pool = w + o; o += (size_t)GG * CC;
    float* cnt  = w + o; o += (size_t)GG;
    size_t total_floats = o;
    if (ws_size < total_floats * sizeof(float)) return;
    unsigned* nmax = (unsigned*)nmaxf;

    const int TB = 256;
    auto cdiv = [](long a, long b) { return (int)((a + b - 1) / b); };

    // Zero everything that accumulates (acc1 .. cnt, one contiguous range)
    k_zero<<<2048, TB, 0, stream>>>(acc1, total_floats - zstart);

    // self-loop mean attrs
    k_deg_attr<<<cdiv(EE, TB), TB, 0, stream>>>(eidx, eattr, deg, mattr);
    k_mean_attr<<<cdiv(NN, TB), TB, 0, stream>>>(deg, mattr);

    // GAT1
    k_lin1<<<cdiv((long)NN * D1, TB), TB, 0, stream>>>(x, Wl1, bl1, Wr1, br1, xl1, xr1);
    k_logits1<<<cdiv((long)EP * 32, TB), TB, 0, stream>>>(eidx, eattr, mattr, xl1, xr1,
                                                          We1, att1, elog, nmax);
    k_expsum1<<<cdiv((long)EP * H1, TB), TB, 0, stream>>>(eidx, nmax, elog, nsum);
    k_agg1<<<cdiv((long)EP * 32, TB), TB, 0, stream>>>(eidx, eattr, mattr, xl1, elog, nsum, acc1);
    k_finish1<<<cdiv((long)NN * D1, TB), TB, 0, stream>>>(acc1, bias1);   // acc1 == h1

    // GAT2 linears via WMMA f32 16x16x4; weights staged to 128KB dynamic LDS
    // with async global->LDS copies (ASYNCcnt path).
    k_lin2_wmma<<<cdiv(3125, 8), TB, 2 * D1 * CC * sizeof(float), stream>>>(
        acc1, Wl2, bl2, Wr2, br2, xl1, xr1);

    // reset segment max/sum for layer 2
    k_zero<<<256, TB, 0, stream>>>(nmaxf, (size_t)NN * H1 * 2);

    k_logits2<<<cdiv((long)EP * 32, TB), TB, 0, stream>>>(eidx, eattr, mattr, xl1, xr1,
                                                          We2, att2, elog, nmax);
    k_expsum2<<<cdiv((long)EP, TB), TB, 0, stream>>>(eidx, nmax, elog, nsum);
    k_agg2<<<cdiv((long)EP * 32, TB), TB, 0, stream>>>(eidx, eattr, mattr, xl1, elog, nsum, acc2);
    k_finish2<<<cdiv((long)NN * CC, TB), TB, 0, stream>>>(acc2, bias2);   // acc2 == h2

    // pool + GRU + classifier
    k_pool<<<cdiv((long)NN * CC, TB), TB, 0, stream>>>(batch, acc2, pool, cnt);
    k_gru_head<<<1, 64, 0, stream>>>(pool, cnt, Wi, Wh, bi, bh, Wc1, bc1, Wc2, bc2, out);
}